// Attention_75402445849147
// MI455X (gfx1250) — compile-verified
//
#include <hip/hip_runtime.h>
#include <math.h>

#define BB   4
#define CCH  256
#define CQK  32
#define NN   16384
#define XPAD 68   // x-tile row pad: bank-conflict-free for stride-68 access
#define MSP  36   // Ms row stride in k_final (36*l mod 64 distinct for l<16)
#define QSP  36   // Qs row stride in k_final

typedef __attribute__((ext_vector_type(2))) float v2f;
typedef __attribute__((ext_vector_type(8))) float v8f;

__device__ __forceinline__ v8f wmma4(v2f a, v2f b, v8f c) {
  // D = A(16x4) * B(4x16) + C,  f32 WMMA
  return __builtin_amdgcn_wmma_f32_16x16x4_f32(false, a, false, b, (short)0, c,
                                               false, false);
}
__device__ __forceinline__ v8f zero8() {
  v8f z = {0.f, 0.f, 0.f, 0.f, 0.f, 0.f, 0.f, 0.f};
  return z;
}

// gfx1250 async global->LDS copy, tracked by ASYNCcnt (no VGPR round-trip).
__device__ __forceinline__ void async_b32(unsigned lds_off, const float* gptr) {
  asm volatile("global_load_async_to_lds_b32 %0, %1, off"
               :: "v"(lds_off), "v"(gptr)
               : "memory");
}
__device__ __forceinline__ void wait_async0() {
  asm volatile("s_wait_asynccnt 0" ::: "memory");
}

// ---------------------------------------------------------------------------
// Kernel 0: zero the small accumulators (matrixT, value_sum, ksum)
// ---------------------------------------------------------------------------
__global__ void k_zero(float* __restrict__ p, int n) {
  int i = blockIdx.x * 256 + threadIdx.x;
  if (i < n) p[i] = 0.f;
}

// ---------------------------------------------------------------------------
// Kernel 1: fused QKV projection (+bias, +L2 norm of Q/K over 32 channels)
// 256 threads (8 waves), 64 positions/block. x staged TRANSPOSED [n][c] in
// LDS via double-buffered async-to-LDS copies, so every WMMA B operand is a
// single ds_load_b64. Wave w owns V c-tiles {2w,2w+1}; waves 0..3 also own
// the Q/K m-tiles. Outputs: V[b][256][N] (=d_out), Qn/Kn [b][N][32].
// ---------------------------------------------------------------------------
__global__ __launch_bounds__(256) void k_qkv(
    const float* __restrict__ x,
    const float* __restrict__ Wq, const float* __restrict__ bq,
    const float* __restrict__ Wk, const float* __restrict__ bk,
    const float* __restrict__ Wv, const float* __restrict__ bv,
    float* __restrict__ V, float* __restrict__ Qn, float* __restrict__ Kn)
{
  __shared__ float xsA[64 * XPAD];  // transposed x chunk: [n(64)][c(64)+pad]
  __shared__ float xsB[64 * XPAD];  // double buffer
  __shared__ float qs [64 * XPAD];  // raw Q rows 0..31 / K rows 32..63: [c][n]

  const int nblk = blockIdx.x;
  const int b    = blockIdx.y;
  const int n0   = nblk * 64;
  const int tid  = threadIdx.x;
  const int lane = tid & 31;
  const int w    = tid >> 5;
  const int lm   = lane & 15;
  const int hi   = lane >> 4;       // 0/1
  const int kb   = hi * 2;          // K-base within 4-wide K slice

  const float* xb = x + (size_t)b * CCH * NN;
  const unsigned baseA = (unsigned)(size_t)(void*)xsA;
  const unsigned baseB = (unsigned)(size_t)(void*)xsB;

  v8f accV[2][4];
  v8f accQ[4];
#pragma unroll
  for (int i = 0; i < 2; ++i)
#pragma unroll
    for (int nt = 0; nt < 4; ++nt) accV[i][nt] = zero8();
#pragma unroll
  for (int nt = 0; nt < 4; ++nt) accQ[nt] = zero8();

  const bool   hasQK = (w < 4);
  const float* Wm    = (w < 2) ? Wq : Wk;   // waves 0,1 -> Q; 2,3 -> K
  const float* bqk   = (w < 2) ? bq : bk;
  const int    qkm   = (w & 1) * 16;        // m-row base inside the 32

  // issue one 64x64 chunk: 128 wave-instructions; wave w issues 16.
  // per instruction: 32 lanes read 128B contiguous of channel c, scatter
  // to LDS column c of the transposed tile (stride 68 -> conflict-free).
  auto issue_chunk = [&](int chunk, unsigned lbase) {
    const float* gb = xb + (size_t)(chunk * 64) * NN + n0;
#pragma unroll 2
    for (int i = 0; i < 16; ++i) {
      const int idx = w * 16 + i;
      const int c   = idx >> 1;
      const int h   = idx & 1;
      const int nl  = h * 32 + lane;
      async_b32(lbase + (unsigned)(nl * XPAD + c) * 4u,
                gb + (size_t)c * NN + nl);
    }
  };

  issue_chunk(0, baseA);
  wait_async0();
  __syncthreads();

#pragma unroll 1
  for (int chunk = 0; chunk < 4; ++chunk) {
    const float* xs = (chunk & 1) ? xsB : xsA;
    if (chunk < 3) issue_chunk(chunk + 1, (chunk & 1) ? baseA : baseB);
    const int cb = chunk * 64;

#pragma unroll 2
    for (int kk = 0; kk < 16; ++kk) {
      const int c4 = kk * 4;
      v2f Bt[4];
#pragma unroll
      for (int nt = 0; nt < 4; ++nt) {
        const float* p = xs + (nt * 16 + lm) * XPAD + c4 + kb;
        Bt[nt][0] = p[0];          // contiguous pair -> one ds_load_b64
        Bt[nt][1] = p[1];
      }
      const int cg = cb + c4 + kb;  // global input-channel for A loads
#pragma unroll
      for (int i = 0; i < 2; ++i) {
        const int mrow = (w * 2 + i) * 16 + lm;
        const float* ap = Wv + (size_t)mrow * CCH + cg;
        v2f A; A[0] = ap[0]; A[1] = ap[1];
#pragma unroll
        for (int nt = 0; nt < 4; ++nt)
          accV[i][nt] = wmma4(A, Bt[nt], accV[i][nt]);
      }
      if (hasQK) {
        const float* ap = Wm + (size_t)(qkm + lm) * CCH + cg;
        v2f A; A[0] = ap[0]; A[1] = ap[1];
#pragma unroll
        for (int nt = 0; nt < 4; ++nt)
          accQ[nt] = wmma4(A, Bt[nt], accQ[nt]);
      }
    }
    wait_async0();     // next buffer's writes complete (this wave)
    __syncthreads();   // all waves done reading cur + writing next
  }

  // ---- V: bias + store to global [b][c][n] ----
#pragma unroll
  for (int i = 0; i < 2; ++i) {
    const int ct = w * 2 + i;
#pragma unroll
    for (int j = 0; j < 8; ++j) {
      const int   crow = ct * 16 + hi * 8 + j;
      const float bias = bv[crow];
#pragma unroll
      for (int nt = 0; nt < 4; ++nt)
        V[((size_t)b * CCH + crow) * NN + n0 + nt * 16 + lm] =
            accV[i][nt][j] + bias;
    }
  }

  // ---- Q/K: bias + write raw tiles to LDS ([c][n] layout) ----
  if (hasQK) {
    const int rbase = ((w >= 2) ? 32 : 0) + qkm;
#pragma unroll
    for (int j = 0; j < 8; ++j) {
      const int   r    = rbase + hi * 8 + j;
      const float bias = bqk[qkm + hi * 8 + j];
#pragma unroll
      for (int nt = 0; nt < 4; ++nt)
        qs[r * XPAD + nt * 16 + lm] = accQ[nt][j] + bias;
    }
  }
  __syncthreads();

  // ---- L2-normalize columns; write Qn/Kn as [b][n][32] ----
  if (tid < 128) {
    const int n   = tid & 63;
    const int isK = tid >> 6;
    const float* col = qs + isK * 32 * XPAD + n;
    float v[32];
    float ss = 0.f;
#pragma unroll
    for (int m = 0; m < 32; ++m) { v[m] = col[m * XPAD]; ss += v[m] * v[m]; }
    const float inv = 1.0f / sqrtf(ss);
    float* dst = (isK ? Kn : Qn) + ((size_t)b * NN + n0 + n) * 32;
#pragma unroll
    for (int m = 0; m < 32; m += 4) {
      float4 o;
      o.x = v[m] * inv;     o.y = v[m + 1] * inv;
      o.z = v[m + 2] * inv; o.w = v[m + 3] * inv;
      *(float4*)(dst + m) = o;
    }
  }
}

// ---------------------------------------------------------------------------
// Kernel 2: matrixT[b][c][m] = sum_n V[c,n]*Kn[n,m];  value_sum; ksum.
// Grid (slab=4, ctile=16, b=4); 8 waves each reduce 512 positions, LDS
// atomic-add reduce, then global f32 atomics.
// ---------------------------------------------------------------------------
__global__ __launch_bounds__(256) void k_matrix(
    const float* __restrict__ V, const float* __restrict__ Kn,
    float* __restrict__ MT, float* __restrict__ vsum, float* __restrict__ ksum)
{
  __shared__ float tile_s[16 * 32];
  __shared__ float vs_s[16];
  __shared__ float ks_s[32];

  const int slab = blockIdx.x, ct = blockIdx.y, b = blockIdx.z;
  const int tid = threadIdx.x, lane = tid & 31, w = tid >> 5;
  const int lm = lane & 15, hi = lane >> 4, kb = hi * 2;

  if (tid < 16) vs_s[tid] = 0.f;
  if (tid < 32) ks_s[tid] = 0.f;
  for (int i = tid; i < 512; i += 256) tile_s[i] = 0.f;
  __syncthreads();

  const int nbase = slab * 4096 + w * 512;
  const float* vrow = V + ((size_t)b * CCH + ct * 16 + lm) * NN;
  const float* knb  = Kn + (size_t)b * NN * 32;

  v8f acc0 = zero8(), acc1 = zero8();
  float rs = 0.f, k0s = 0.f, k1s = 0.f;
#pragma unroll 4
  for (int n4 = 0; n4 < 512; n4 += 4) {
    const int n = nbase + n4 + kb;
    v2f A; A[0] = vrow[n]; A[1] = vrow[n + 1];
    rs += A[0] + A[1];
    const float* kp = knb + (size_t)n * 32 + lm;
    v2f B0, B1;
    B0[0] = kp[0];  B0[1] = kp[32];
    B1[0] = kp[16]; B1[1] = kp[48];
    k0s += B0[0] + B0[1];
    k1s += B1[0] + B1[1];
    acc0 = wmma4(A, B0, acc0);
    acc1 = wmma4(A, B1, acc1);
  }

#pragma unroll
  for (int j = 0; j < 8; ++j) {
    const int row = hi * 8 + j;
    atomicAdd(&tile_s[row * 32 + lm], acc0[j]);
    atomicAdd(&tile_s[row * 32 + 16 + lm], acc1[j]);
  }
  atomicAdd(&vs_s[lm], rs);
  if (ct == 0) {
    atomicAdd(&ks_s[lm], k0s);
    atomicAdd(&ks_s[16 + lm], k1s);
  }
  __syncthreads();

  for (int i = tid; i < 512; i += 256)
    atomicAdd(&MT[((size_t)b * CCH + ct * 16) * 32 + i], tile_s[i]);
  if (tid < 16) atomicAdd(&vsum[b * CCH + ct * 16 + tid], vs_s[tid]);
  if (ct == 0 && tid < 32) atomicAdd(&ksum[b * CQK + tid], ks_s[tid]);
}

// ---------------------------------------------------------------------------
// Kernel 3: out[b][c][n] = tailor(n) * (vsum[c] + sum_m MT[c][m]*Qn[n][m])
// with tailor(n) = gamma / (N + Qn[n,:] . (ksum + eps)).
// Grid (N/128, B); matrixT + Qn tile + vsum staged in LDS with padded row
// strides (36) so every ds_load_b64 of A/B is bank-conflict-free.
// ---------------------------------------------------------------------------
__global__ __launch_bounds__(256) void k_final(
    const float* __restrict__ Qn, const float* __restrict__ MT,
    const float* __restrict__ vsum, const float* __restrict__ ksum,
    const float* __restrict__ gamma, float* __restrict__ out)
{
  __shared__ float Ms[CCH * MSP];   // 36 KB
  __shared__ float Qs[128 * QSP];   // 18 KB
  __shared__ float vs_s[CCH];
  __shared__ float tg_s[128];

  const int nblk = blockIdx.x, b = blockIdx.y;
  const int n0 = nblk * 128;
  const int tid = threadIdx.x, lane = tid & 31, w = tid >> 5;
  const int lm = lane & 15, hi = lane >> 4, kb = hi * 2;

  {
    // Ms: thread t stages matrixT row t (8 float4, padded dst stride)
    const float4* src = (const float4*)(MT + (size_t)b * CCH * 32 + tid * 32);
    float4*       dst = (float4*)(Ms + tid * MSP);
#pragma unroll
    for (int i = 0; i < 8; ++i) dst[i] = src[i];
    // Qs: two threads per row of 32 floats
    const int qr = tid >> 1, qseg = (tid & 1) * 4;
    const float4* qsrc =
        (const float4*)(Qn + ((size_t)b * NN + n0 + qr) * 32) + qseg;
    float4* qdst = (float4*)(Qs + qr * QSP) + qseg;
#pragma unroll
    for (int i = 0; i < 4; ++i) qdst[i] = qsrc[i];
    vs_s[tid] = vsum[b * CCH + tid];
  }
  __syncthreads();

  if (tid < 128) {
    const float* q = Qs + tid * QSP;
    float dot = 0.f;
#pragma unroll
    for (int m = 0; m < 32; ++m) dot += q[m] * (ksum[b * CQK + m] + 1e-6f);
    tg_s[tid] = gamma[0] / (16384.0f + dot);
  }
  __syncthreads();

#pragma unroll 1
  for (int i = 0; i < 2; ++i) {
    const int ct = w * 2 + i;
    v2f A[8];
    {
      const float* ap = Ms + (ct * 16 + lm) * MSP + kb;
#pragma unroll
      for (int k = 0; k < 8; ++k) { A[k][0] = ap[k * 4]; A[k][1] = ap[k * 4 + 1]; }
    }
    float vinit[8];
#pragma unroll
    for (int j = 0; j < 8; ++j) vinit[j] = vs_s[ct * 16 + hi * 8 + j];

#pragma unroll 1
    for (int nt = 0; nt < 8; ++nt) {
      v8f acc;
#pragma unroll
      for (int j = 0; j < 8; ++j) acc[j] = vinit[j];
      const float* qp = Qs + (nt * 16 + lm) * QSP + kb;
#pragma unroll
      for (int k = 0; k < 8; ++k) {
        v2f Bv; Bv[0] = qp[k * 4]; Bv[1] = qp[k * 4 + 1];
        acc = wmma4(A[k], Bv, acc);
      }
      const float tg = tg_s[nt * 16 + lm];
#pragma unroll
      for (int j = 0; j < 8; ++j) {
        const int crow = ct * 16 + hi * 8 + j;
        out[((size_t)b * CCH + crow) * NN + n0 + nt * 16 + lm] = acc[j] * tg;
      }
    }
  }
}

// ---------------------------------------------------------------------------
extern "C" void kernel_launch(void* const* d_in, const int* in_sizes, int n_in,
                              void* d_out, int out_size, void* d_ws, size_t ws_size,
                              hipStream_t stream)
{
  (void)in_sizes; (void)n_in; (void)out_size; (void)ws_size;
  const float* x     = (const float*)d_in[0];
  const float* Wq    = (const float*)d_in[1];
  const float* bq    = (const float*)d_in[2];
  const float* Wk    = (const float*)d_in[3];
  const float* bk    = (const float*)d_in[4];
  const float* Wv    = (const float*)d_in[5];
  const float* bv    = (const float*)d_in[6];
  const float* gamma = (const float*)d_in[7];
  float* out = (float*)d_out;

  // workspace layout (floats): Qn | Kn | MT | vsum | ksum
  float* Qn   = (float*)d_ws;                       // B*N*32
  float* Kn   = Qn + (size_t)BB * NN * 32;          // B*N*32
  float* MT   = Kn + (size_t)BB * NN * 32;          // B*256*32
  float* vsum = MT + (size_t)BB * CCH * 32;         // B*256
  float* ksum = vsum + (size_t)BB * CCH;            // B*32
  float* V    = out;  // reuse d_out as V scratch; k_final fully overwrites it

  const int accN = BB * CCH * 32 + BB * CCH + BB * CQK;  // 33920
  k_zero<<<dim3((accN + 255) / 256), 256, 0, stream>>>(MT, accN);
  k_qkv<<<dim3(NN / 64, BB), 256, 0, stream>>>(x, Wq, bq, Wk, bk, Wv, bv,
                                               V, Qn, Kn);
  k_matrix<<<dim3(4, 16, BB), 256, 0, stream>>>(V, Kn, MT, vsum, ksum);
  k_final<<<dim3(NN / 128, BB), 256, 0, stream>>>(Qn, MT, vsum, ksum, gamma, out);
}